// Router_38482906972898
// MI455X (gfx1250) — compile-verified
//
#include <hip/hip_runtime.h>
#include <hip/hip_bf16.h>
#include <math.h>

typedef __attribute__((ext_vector_type(2))) float v2f;
typedef __attribute__((ext_vector_type(4))) float v4f;
typedef __attribute__((ext_vector_type(8))) float v8f;

#define HID   2048
#define NEXP  64
#define NTOK  16384   // 4 * 4096
#define TOPK  4

// Workspace layout (in floats):
#define WS_LOGITS 0
#define WS_PROB   (NTOK * NEXP)        // 64 floats: sum of softmax probs per expert
#define WS_CNT    (WS_PROB + NEXP)     // 64 floats: top-k assignment counts per expert
#define WS_Z      (WS_CNT + NEXP)      // 1 float : sum of logsumexp^2

// ---------------------------------------------------------------------------
// Kernel 0: zero the reduction accumulators (must happen every call).
// ---------------------------------------------------------------------------
__global__ __launch_bounds__(256) void k_init(float* __restrict__ ws) {
  int i = threadIdx.x;
  if (i < NEXP) {
    ws[WS_PROB + i] = 0.0f;
    ws[WS_CNT + i]  = 0.0f;
  }
  if (i == 0) ws[WS_Z] = 0.0f;
}

// ---------------------------------------------------------------------------
// Kernel 1: gate GEMM via V_WMMA_F32_16X16X4_F32.
// Each wave32 computes a 16-token x 64-expert tile; K loop over 2048 in
// steps of 4. A fragment: hidden[tok0 + lane%16][k + 2*(lane/16) + {0,1}].
// B fragment (n-tile j): gate[j*16 + lane%16][k + 2*(lane/16) + {0,1}].
// Output clamped to [-10,10] and stored row-major [token][expert].
// ---------------------------------------------------------------------------
__global__ __launch_bounds__(256) void k_gemm(const float* __restrict__ hid,
                                              const float* __restrict__ gate,
                                              float* __restrict__ logits) {
  const int lane = threadIdx.x & 31;
  const int wave = threadIdx.x >> 5;
  const int half = lane >> 4;       // 0: K+0..1, 1: K+2..3
  const int l16  = lane & 15;
  const int tok0 = blockIdx.x * (8 * 16) + wave * 16;

  const float* ap  = hid  + (size_t)(tok0 + l16) * HID + 2 * half;
  const float* bp0 = gate + (size_t)( 0 + l16) * HID + 2 * half;
  const float* bp1 = gate + (size_t)(16 + l16) * HID + 2 * half;
  const float* bp2 = gate + (size_t)(32 + l16) * HID + 2 * half;
  const float* bp3 = gate + (size_t)(48 + l16) * HID + 2 * half;

  v8f acc0 = {}, acc1 = {}, acc2 = {}, acc3 = {};

#pragma unroll 4
  for (int k = 0; k < HID; k += 4) {
    v2f a  = *(const v2f*)(ap  + k);
    v2f b0 = *(const v2f*)(bp0 + k);
    v2f b1 = *(const v2f*)(bp1 + k);
    v2f b2 = *(const v2f*)(bp2 + k);
    v2f b3 = *(const v2f*)(bp3 + k);
    acc0 = __builtin_amdgcn_wmma_f32_16x16x4_f32(false, a, false, b0, (short)0, acc0, false, false);
    acc1 = __builtin_amdgcn_wmma_f32_16x16x4_f32(false, a, false, b1, (short)0, acc1, false, false);
    acc2 = __builtin_amdgcn_wmma_f32_16x16x4_f32(false, a, false, b2, (short)0, acc2, false, false);
    acc3 = __builtin_amdgcn_wmma_f32_16x16x4_f32(false, a, false, b3, (short)0, acc3, false, false);
  }

  // C/D layout: VGPR r of lane L holds M = r + 8*(L>=16), N = L%16.
#pragma unroll
  for (int r = 0; r < 8; ++r) {
    size_t row = (size_t)(tok0 + r + 8 * half) * NEXP;
    float v0 = fminf(10.0f, fmaxf(-10.0f, acc0[r]));
    float v1 = fminf(10.0f, fmaxf(-10.0f, acc1[r]));
    float v2 = fminf(10.0f, fmaxf(-10.0f, acc2[r]));
    float v3 = fminf(10.0f, fmaxf(-10.0f, acc3[r]));
    logits[row +  0 + l16] = v0;
    logits[row + 16 + l16] = v1;
    logits[row + 32 + l16] = v2;
    logits[row + 48 + l16] = v3;
  }
}

// ---------------------------------------------------------------------------
// Kernel 2: per-token routing. One thread per token.
// ---------------------------------------------------------------------------
__global__ __launch_bounds__(256) void k_route(const float* __restrict__ logits,
                                               float* __restrict__ out,
                                               float* __restrict__ ws) {
  __shared__ float s_prob[NEXP];
  __shared__ float s_cnt[NEXP];
  __shared__ float s_z[1];
  const int tid = threadIdx.x;
  if (tid < NEXP) { s_prob[tid] = 0.0f; s_cnt[tid] = 0.0f; }
  if (tid == 0) s_z[0] = 0.0f;
  __syncthreads();

  const int token = blockIdx.x * 256 + tid;
  float l[NEXP];
  const v4f* lp = (const v4f*)(logits + (size_t)token * NEXP);
#pragma unroll
  for (int i = 0; i < NEXP / 4; ++i) {
    v4f t = lp[i];
    l[4 * i + 0] = t[0];
    l[4 * i + 1] = t[1];
    l[4 * i + 2] = t[2];
    l[4 * i + 3] = t[3];
  }

  // Full softmax stats over all 64 experts (for lb-loss and z-loss).
  float m = l[0];
#pragma unroll
  for (int e = 1; e < NEXP; ++e) m = fmaxf(m, l[e]);
  float s = 0.0f;
#pragma unroll
  for (int e = 0; e < NEXP; ++e) s += __expf(l[e] - m);
  const float inv = 1.0f / s;
  const float logz = m + __logf(s);
#pragma unroll
  for (int e = 0; e < NEXP; ++e) atomicAdd(&s_prob[e], __expf(l[e] - m) * inv);

  // Top-4 (first-index wins on ties, matching jax.lax.top_k).
  int   c0, c1, c2, c3;
  float v0, v1, v2, v3;
  {
    float best = -3.0e38f; int bi = 0;
#pragma unroll
    for (int e = 0; e < NEXP; ++e) { if (l[e] > best) { best = l[e]; bi = e; } }
    c0 = bi; v0 = best;
  }
  {
    float best = -3.0e38f; int bi = 0;
#pragma unroll
    for (int e = 0; e < NEXP; ++e) {
      float v = (e == c0) ? -3.0e38f : l[e];
      if (v > best) { best = v; bi = e; }
    }
    c1 = bi; v1 = best;
  }
  {
    float best = -3.0e38f; int bi = 0;
#pragma unroll
    for (int e = 0; e < NEXP; ++e) {
      float v = (e == c0 || e == c1) ? -3.0e38f : l[e];
      if (v > best) { best = v; bi = e; }
    }
    c2 = bi; v2 = best;
  }
  {
    float best = -3.0e38f; int bi = 0;
#pragma unroll
    for (int e = 0; e < NEXP; ++e) {
      float v = (e == c0 || e == c1 || e == c2) ? -3.0e38f : l[e];
      if (v > best) { best = v; bi = e; }
    }
    c3 = bi; v3 = best;
  }

  // Softmax over the 4 selected (v0 is the max).
  float e0 = 1.0f;
  float e1 = __expf(v1 - v0);
  float e2 = __expf(v2 - v0);
  float e3 = __expf(v3 - v0);
  float rn = 1.0f / (e0 + e1 + e2 + e3);

  float* wout = out + (size_t)token * TOPK;
  float* eout = out + (size_t)NTOK * TOPK + (size_t)token * TOPK;
  wout[0] = e0 * rn; wout[1] = e1 * rn; wout[2] = e2 * rn; wout[3] = e3 * rn;
  eout[0] = (float)c0; eout[1] = (float)c1; eout[2] = (float)c2; eout[3] = (float)c3;

  atomicAdd(&s_cnt[c0], 1.0f);
  atomicAdd(&s_cnt[c1], 1.0f);
  atomicAdd(&s_cnt[c2], 1.0f);
  atomicAdd(&s_cnt[c3], 1.0f);
  atomicAdd(&s_z[0], logz * logz);

  __syncthreads();
  if (tid < NEXP) {
    atomicAdd(&ws[WS_PROB + tid], s_prob[tid]);
    atomicAdd(&ws[WS_CNT + tid], s_cnt[tid]);
  }
  if (tid == 0) atomicAdd(&ws[WS_Z], s_z[0]);
}

// ---------------------------------------------------------------------------
// Kernel 3: finalize the two scalar losses.
// ---------------------------------------------------------------------------
__global__ __launch_bounds__(64) void k_final(const float* __restrict__ ws,
                                              float* __restrict__ out) {
  __shared__ float s[NEXP];
  const int t = threadIdx.x;
  const float invT = 1.0f / (float)NTOK;
  s[t] = (ws[WS_CNT + t] * invT) * (ws[WS_PROB + t] * invT);
  __syncthreads();
  if (t == 0) {
    float acc = 0.0f;
    for (int i = 0; i < NEXP; ++i) acc += s[i];
    out[2 * NTOK * TOPK + 0] = (float)NEXP * acc * 0.01f;          // lb_loss
    out[2 * NTOK * TOPK + 1] = (ws[WS_Z] * invT) * 0.01f;          // z_loss
  }
}

// ---------------------------------------------------------------------------
extern "C" void kernel_launch(void* const* d_in, const int* in_sizes, int n_in,
                              void* d_out, int out_size, void* d_ws, size_t ws_size,
                              hipStream_t stream) {
  const float* hid  = (const float*)d_in[0];   // [4,4096,2048] fp32
  const float* gate = (const float*)d_in[1];   // [64,2048] fp32
  float* out = (float*)d_out;                  // 65536 + 65536 + 2 floats
  float* ws  = (float*)d_ws;                   // >= (NTOK*NEXP + 129) floats

  k_init <<<1, 256, 0, stream>>>(ws);
  k_gemm <<<NTOK / 128, 256, 0, stream>>>(hid, gate, ws + WS_LOGITS);
  k_route<<<NTOK / 256, 256, 0, stream>>>(ws + WS_LOGITS, out, ws);
  k_final<<<1, 64, 0, stream>>>(ws, out);
}